// GeoIE_past_77214922047875
// MI455X (gfx1250) — compile-verified
//
#include <hip/hip_runtime.h>

typedef __attribute__((ext_vector_type(2))) float v2f;
typedef __attribute__((ext_vector_type(8))) float v8f;

#define NB   4096   // B
#define NH   50     // H
#define NNEG 10     // N
#define ND   64     // D

#define A_STRIDE 68   // 68*4B = 272B rows -> conflict-free b64 fragment loads
#define B_STRIDE 68
#define D_STRIDE 33

// ---- CDNA5 async HBM->LDS path (ASYNCcnt), with safe fallback -----------------
#if __has_builtin(__builtin_amdgcn_global_load_async_to_lds_b128) && \
    __has_builtin(__builtin_amdgcn_s_wait_asynccnt)
#define USE_ASYNC_LDS 1
#else
#define USE_ASYNC_LDS 0
#endif

#if USE_ASYNC_LDS
// Builtin signature (from hipcc diagnostic): param0 is a global-AS pointer to a
// 16-byte int vector (non-const); param1 is the LDS-side pointer.
typedef __attribute__((vector_size(4 * sizeof(int)))) int v4i_vs;
typedef __attribute__((address_space(1))) v4i_vs* gas_v4i_ptr;
typedef __attribute__((address_space(3))) v4i_vs* las_v4i_ptr;

__device__ __forceinline__ void async_cp16(const void* g, void* l) {
    // global_load_async_to_lds_b128: per-lane global addr -> per-lane LDS addr
    __builtin_amdgcn_global_load_async_to_lds_b128(
        (gas_v4i_ptr)g, (las_v4i_ptr)l, /*offset=*/0, /*cpol=*/0);
}
#endif

__global__ __launch_bounds__(256) void geoie_fused_kernel(
    const float* __restrict__ cuj,
    const int*   __restrict__ user_id,
    const int*   __restrict__ target,
    const int*   __restrict__ neg_p,
    const int*   __restrict__ History,
    const float* __restrict__ distance,
    const float* __restrict__ ng_distance,
    const float* __restrict__ a_ptr,
    const float* __restrict__ b_ptr,
    const float* __restrict__ UserPreference,
    const float* __restrict__ PoiPreference,
    const float* __restrict__ GeoInfluence,
    const float* __restrict__ GeoSusceptibility,
    float* __restrict__ out)
{
    __shared__ float Alds[64 * A_STRIDE];   // rows 0..49 = g[h], row 50 = UPre, 51..63 = 0
    __shared__ float Blds[32 * B_STRIDE];   // B columns as rows: 0 = hj, 1..10 = Neghj[n],
                                            //                    16 = PPre, 17..26 = NegPPre[n]
    __shared__ float Dlds[64 * D_STRIDE];   // D = A*B scratch

    const int b   = blockIdx.x;
    const int tid = threadIdx.x;

    // ---------------- stage A into LDS (64 rows x 16 float4 chunks) ----------------
    for (int i = tid; i < 64 * 16; i += 256) {
        const int row = i >> 4;
        const int c4  = (i & 15) << 2;
        float* dst = &Alds[row * A_STRIDE + c4];
#if USE_ASYNC_LDS
        if (row < NH) {
            async_cp16(&GeoInfluence[(size_t)History[b * NH + row] * ND + c4], dst);
        } else if (row == NH) {
            async_cp16(&UserPreference[(size_t)user_id[b] * ND + c4], dst);
        } else {
            *(float4*)dst = make_float4(0.f, 0.f, 0.f, 0.f);
        }
#else
        float4 v = make_float4(0.f, 0.f, 0.f, 0.f);
        if (row < NH) {
            v = *(const float4*)&GeoInfluence[(size_t)History[b * NH + row] * ND + c4];
        } else if (row == NH) {
            v = *(const float4*)&UserPreference[(size_t)user_id[b] * ND + c4];
        }
        *(float4*)dst = v;
#endif
    }
    // ---------------- stage B (column-major) into LDS (32 rows x 16 chunks) --------
    for (int i = tid; i < 32 * 16; i += 256) {
        const int row = i >> 4;
        const int c4  = (i & 15) << 2;
        float* dst = &Blds[row * B_STRIDE + c4];
#if USE_ASYNC_LDS
        if (row == 0) {
            async_cp16(&GeoSusceptibility[(size_t)target[b] * ND + c4], dst);
        } else if (row <= 10) {
            async_cp16(&GeoSusceptibility[(size_t)neg_p[b * NNEG + (row - 1)] * ND + c4], dst);
        } else if (row == 16) {
            async_cp16(&PoiPreference[(size_t)target[b] * ND + c4], dst);
        } else if (row >= 17 && row <= 26) {
            async_cp16(&PoiPreference[(size_t)neg_p[b * NNEG + (row - 17)] * ND + c4], dst);
        } else {
            *(float4*)dst = make_float4(0.f, 0.f, 0.f, 0.f);
        }
#else
        float4 v = make_float4(0.f, 0.f, 0.f, 0.f);
        if (row == 0) {
            v = *(const float4*)&GeoSusceptibility[(size_t)target[b] * ND + c4];
        } else if (row <= 10) {
            v = *(const float4*)&GeoSusceptibility[(size_t)neg_p[b * NNEG + (row - 1)] * ND + c4];
        } else if (row == 16) {
            v = *(const float4*)&PoiPreference[(size_t)target[b] * ND + c4];
        } else if (row >= 17 && row <= 26) {
            v = *(const float4*)&PoiPreference[(size_t)neg_p[b * NNEG + (row - 17)] * ND + c4];
        }
        *(float4*)dst = v;
#endif
    }
#if USE_ASYNC_LDS
    __builtin_amdgcn_s_wait_asynccnt(0);   // all async HBM->LDS transfers landed
#endif
    __syncthreads();

    // ---------------- WMMA: D[64x32] = A[64x64] x B[64x32], fp32 -------------------
    // 8 waves, one 16x16 tile each: mi = wave>>1 (4 M-tiles), ni = wave&1 (2 N-tiles)
    const int wave = tid >> 5;
    const int lane = tid & 31;
    const int m0 = (wave >> 1) << 4;
    const int n0 = (wave & 1) << 4;
    const int mr = m0 + (lane & 15);
    const int nr = n0 + (lane & 15);
    const int kh = (lane >> 4) << 1;     // 0 for lanes 0-15, 2 for lanes 16-31

    v8f c = {};
    #pragma unroll
    for (int k0 = 0; k0 < 64; k0 += 4) {
        v2f av = *(const v2f*)&Alds[mr * A_STRIDE + k0 + kh];
        v2f bv = *(const v2f*)&Blds[nr * B_STRIDE + k0 + kh];
        // D = A*B + C, full fp32 (V_WMMA_F32_16X16X4_F32)
        c = __builtin_amdgcn_wmma_f32_16x16x4_f32(
                /*neg_a=*/false, av, /*neg_b=*/false, bv,
                /*c_mod=*/(short)0, c, /*reuse_a=*/false, /*reuse_b=*/false);
    }
    #pragma unroll
    for (int j = 0; j < 8; ++j) {
        const int m = m0 + j + ((lane >> 4) << 3);
        const int n = n0 + (lane & 15);
        Dlds[m * D_STRIDE + n] = c[j];
    }
    __syncthreads();

    // ---------------- epilogue: weighted geo sums + log-sigmoid (wave 0) ----------
    if (wave == 0) {
        const float a_s  = a_ptr[0];
        const float bexp = b_ptr[0];
        float Lsum = 0.f;
        #pragma unroll 1
        for (int j = 0; j < 11; ++j) {   // j==0: positive; j>=1: negative sample j-1
            float partial = 0.f;
            for (int h = lane; h < NH; h += 32) {
                const float d = (j == 0) ? distance[b * NH + h]
                                         : ng_distance[(b * NNEG + (j - 1)) * NH + h];
                const float w = a_s * __powf(d, bexp);       // f = a * d^b
                partial += w * Dlds[h * D_STRIDE + j];       // <g[h], hj/Neghj[n]>
            }
            #pragma unroll
            for (int off = 16; off; off >>= 1)
                partial += __shfl_xor(partial, off, 32);
            const float yij = partial * (1.0f / (float)NH);
            const float tz  = Dlds[NH * D_STRIDE + 16 + j];  // UPre . (P/NegP)Pre
            const float s   = tz + yij;
            const float x   = (j == 0) ? s : -s;             // log_sigmoid(+s) / log_sigmoid(-s)
            Lsum += fminf(x, 0.f) - log1pf(__expf(-fabsf(x)));
        }
        if (lane == 0) {
            const float wuj = 1.0f + log1pf(cuj[b] * 1.0e10f);  // 10**SCALING
            out[b] = -wuj * Lsum;
        }
    }
}

extern "C" void kernel_launch(void* const* d_in, const int* in_sizes, int n_in,
                              void* d_out, int out_size, void* d_ws, size_t ws_size,
                              hipStream_t stream) {
    (void)in_sizes; (void)n_in; (void)out_size; (void)d_ws; (void)ws_size;
    geoie_fused_kernel<<<NB, 256, 0, stream>>>(
        (const float*)d_in[0],   // cuj            [B]
        (const int*)  d_in[1],   // user_id        [B]
        (const int*)  d_in[2],   // target         [B]
        (const int*)  d_in[3],   // neg_p          [B,N]
        (const int*)  d_in[4],   // History        [B,H]
        (const float*)d_in[5],   // distance       [B,H]
        (const float*)d_in[6],   // ng_distance    [B,N,H]
        (const float*)d_in[7],   // a              [1]
        (const float*)d_in[8],   // b              [1]
        (const float*)d_in[9],   // UserPreference [V,D]
        (const float*)d_in[10],  // PoiPreference  [V,D]
        (const float*)d_in[11],  // GeoInfluence   [V,D]
        (const float*)d_in[12],  // GeoSusceptibility [V,D]
        (float*)d_out);          // out            [B]
}